// ZNCC_88467736363637
// MI455X (gfx1250) — compile-verified
//
#include <hip/hip_runtime.h>
#include <stdint.h>

// ---------------------------------------------------------------------------
// ZNCC (5x5 box, zero-pad SAME) for [16,3,512,512] f32 -> [16,1,512,512] f32.
//
// MI455X reasoning: ~117 MB of traffic -> ~5us at 23.3 TB/s, and ~1 GFLOP of
// separable-filter math, so the kernel is near the bw/compute balance point.
// Strategy:
//   * async global->LDS (ASYNCcnt path) to stage 40x40 haloed tiles
//   * VALU separable horizontal 5-tap passes
//   * vertical 5-tap box filter of the 3 product fields as banded matmuls on
//     V_WMMA_F32_16X16X4_F32 (D = V * Hp, V = constant band matrix with the
//     1/25 normalization baked in; K=20 -> 5 chained WMMAs per 16x16 tile)
//   * raw v_sqrt_f32 / v_rcp_f32 for the ncc tail (skip IEEE fixup bloat)
// ---------------------------------------------------------------------------

#define TS   32          // output tile
#define IN   40          // raw tile (TS + 2*4 halo)
#define CS   36          // centered-field tile (TS + 2*2)
#define RSTR 44          // raw LDS row stride (16B-aligned, conflict pad)
#define HSTR 40          // staging LDS row stride
#define EPSF 1e-8f

typedef __attribute__((ext_vector_type(2))) float v2f;
typedef __attribute__((ext_vector_type(4))) float v4f;
typedef __attribute__((ext_vector_type(8))) float v8f;
typedef __attribute__((ext_vector_type(4))) int   v4i;

typedef __attribute__((address_space(1))) v4i* gv4i_p;
typedef __attribute__((address_space(3))) v4i* lv4i_p;

#if __has_builtin(__builtin_amdgcn_global_load_async_to_lds_b128)
#define HAVE_ASYNC_LDS 1
#endif

__device__ __forceinline__ void copy16_to_lds(const float* g, float* l) {
#ifdef HAVE_ASYNC_LDS
  __builtin_amdgcn_global_load_async_to_lds_b128(
      (gv4i_p)(uintptr_t)g,
      (lv4i_p)(uint32_t)(uintptr_t)l,
      0, 0);
#else
  *(v4f*)l = *(const v4f*)g;
#endif
}

__device__ __forceinline__ void wait_async_lds() {
#ifdef HAVE_ASYNC_LDS
#if __has_builtin(__builtin_amdgcn_s_wait_asynccnt)
  __builtin_amdgcn_s_wait_asynccnt(0);
#else
  asm volatile("s_wait_asynccnt 0" ::: "memory");
#endif
#endif
}

__global__ __launch_bounds__(256)
void ZNCC_88467736363637_kernel(const float* __restrict__ x,
                                const float* __restrict__ y,
                                float* __restrict__ out) {
  __shared__ float rawx[IN * RSTR];
  __shared__ float rawy[IN * RSTR];
  __shared__ float hx[IN * HSTR];        // horizontal 5-sums of raw
  __shared__ float hy[IN * HSTR];
  __shared__ float xcx[CS * HSTR];       // centered fields
  __shared__ float xcy[CS * HSTR];
  __shared__ float hp[3][CS * HSTR];     // horizontal 5-sums of products

  const int tid  = threadIdx.x;
  const int lane = tid & 31;
  const int wave = tid >> 5;
  const int ox = blockIdx.x * TS;
  const int oy = blockIdx.y * TS;
  const int b  = blockIdx.z;

  const long plane = 512l * 512l;
  const float* xb = x + (long)b * 3 * plane;
  const float* yb = y + (long)b * 3 * plane;

  float acc[8];
  if (wave < 4) {
#pragma unroll
    for (int i = 0; i < 8; ++i) acc[i] = 0.0f;
  }

  for (int c = 0; c < 3; ++c) {
    const float* xp = xb + (long)c * plane;
    const float* yp = yb + (long)c * plane;

    // ---- Phase 1: stage 40x40 raw tiles (zero pad OOB). W=512 is a multiple
    // of 4 so each 16B group is entirely in or out of bounds.
    for (int t = tid; t < 2 * IN * 10; t += 256) {
      const int f   = t / (IN * 10);
      const int rem = t % (IN * 10);
      const int r   = rem / 10;
      const int g4  = (rem % 10) * 4;
      const int iy  = oy - 4 + r;
      const int gx  = ox - 4 + g4;
      float* dst = (f ? rawy : rawx) + r * RSTR + g4;
      if (iy >= 0 && iy < 512 && gx >= 0 && gx < 512) {
        const float* src = (f ? yp : xp) + (long)iy * 512 + gx;
        copy16_to_lds(src, dst);
      } else {
        dst[0] = 0.0f; dst[1] = 0.0f; dst[2] = 0.0f; dst[3] = 0.0f;
      }
    }
    wait_async_lds();
    __syncthreads();

    // ---- Phase 2: horizontal 5-sums of raw x,y (rows 0..39, cols 0..35).
    for (int t = tid; t < 2 * IN * CS; t += 256) {
      const int f   = t / (IN * CS);
      const int rem = t % (IN * CS);
      const int r   = rem / CS;
      const int cc  = rem % CS;
      const float* rp = (f ? rawy : rawx) + r * RSTR + cc;
      (f ? hy : hx)[r * HSTR + cc] = rp[0] + rp[1] + rp[2] + rp[3] + rp[4];
    }
    __syncthreads();

    // ---- Phase 3: centered fields on 36x36: xc = raw - boxmean(raw).
    for (int t = tid; t < 2 * CS * CS; t += 256) {
      const int f   = t / (CS * CS);
      const int rem = t % (CS * CS);
      const int r   = rem / CS;
      const int cc  = rem % CS;
      const float* hh = (f ? hy : hx) + r * HSTR + cc;
      const float m = (hh[0] + hh[HSTR] + hh[2 * HSTR] + hh[3 * HSTR] + hh[4 * HSTR]) * 0.04f;
      const float v = (f ? rawy : rawx)[(r + 2) * RSTR + (cc + 2)] - m;
      (f ? xcy : xcx)[r * HSTR + cc] = v;
    }
    __syncthreads();

    // ---- Phase 4: horizontal 5-sums of products xx/yy/xy (36 rows x 32 cols).
    for (int t = tid; t < 3 * CS * TS; t += 256) {
      const int f   = t / (CS * TS);
      const int rem = t % (CS * TS);
      const int r   = rem / TS;
      const int cc  = rem % TS;
      const float* ax = xcx + r * HSTR + cc;
      const float* ay = xcy + r * HSTR + cc;
      float s;
      if (f == 0) {
        s = ax[0]*ax[0] + ax[1]*ax[1] + ax[2]*ax[2] + ax[3]*ax[3] + ax[4]*ax[4];
      } else if (f == 1) {
        s = ay[0]*ay[0] + ay[1]*ay[1] + ay[2]*ay[2] + ay[3]*ay[3] + ay[4]*ay[4];
      } else {
        s = ax[0]*ay[0] + ax[1]*ay[1] + ax[2]*ay[2] + ax[3]*ay[3] + ax[4]*ay[4];
      }
      hp[f][r * HSTR + cc] = s;
    }
    __syncthreads();

    // ---- Phase 5: vertical 5-tap via WMMA. D(16x16) = V(16x20) * Hp(20x16),
    // V[m][k] = 0.04 iff m <= k <= m+4 (1/25 baked in). 5 chained k=4 WMMAs.
    // Waves 0..3 each own one 16x16 output quadrant (EXEC all-ones inside).
    if (wave < 4) {
      const int R  = (wave >> 1) * 16;   // output row base
      const int Cc = (wave & 1) * 16;    // output col base
      const int N    = lane & 15;        // A: M index, B/C/D: N index
      const int half = lane >> 4;

      v8f sres[3];
#pragma unroll
      for (int f = 0; f < 3; ++f) {
        v8f s = {};
#pragma unroll
        for (int g = 0; g < 5; ++g) {
          const int k0 = 4 * g + 2 * half;   // this lane's K pair within block
          v2f a, bb;
          a.x = (k0     >= N && k0     <= N + 4) ? 0.04f : 0.0f;
          a.y = (k0 + 1 >= N && k0 + 1 <= N + 4) ? 0.04f : 0.0f;
          bb.x = hp[f][(R + k0    ) * HSTR + (Cc + N)];
          bb.y = hp[f][(R + k0 + 1) * HSTR + (Cc + N)];
          s = __builtin_amdgcn_wmma_f32_16x16x4_f32(
              false, a, false, bb, (short)0, s, false, false);
        }
        sres[f] = s;
      }
#pragma unroll
      for (int i = 0; i < 8; ++i) {
        const float axx = sres[0][i];
        const float ayy = sres[1][i];
        const float axy = sres[2][i];
        // raw v_sqrt_f32 / v_rcp_f32 (~1 ulp): inputs are box sums of squares,
        // O(1) and non-denormal, so skip the IEEE denorm-rescue expansions.
        const float den = __builtin_amdgcn_sqrtf(axx) * __builtin_amdgcn_sqrtf(ayy) + EPSF;
        acc[i] += axy * __builtin_amdgcn_rcpf(den);
      }
    }
    __syncthreads();   // before next channel overwrites LDS
  }

  // ---- Writeout: channel mean. D layout: VGPR i -> row = i + 8*half.
  if (wave < 4) {
    const int R    = (wave >> 1) * 16;
    const int Cc   = (wave & 1) * 16;
    const int N    = lane & 15;
    const int half = lane >> 4;
#pragma unroll
    for (int i = 0; i < 8; ++i) {
      const int row = R + i + 8 * half;
      const int col = Cc + N;
      out[(long)b * plane + (long)(oy + row) * 512 + (ox + col)] =
          acc[i] * (1.0f / 3.0f);
    }
  }
}

extern "C" void kernel_launch(void* const* d_in, const int* in_sizes, int n_in,
                              void* d_out, int out_size, void* d_ws, size_t ws_size,
                              hipStream_t stream) {
  const float* x = (const float*)d_in[0];
  const float* y = (const float*)d_in[1];
  float* out = (float*)d_out;
  dim3 grid(512 / TS, 512 / TS, 16);
  ZNCC_88467736363637_kernel<<<grid, dim3(256), 0, stream>>>(x, y, out);
}